// BCGrounder_29394756174278
// MI455X (gfx1250) — compile-verified
//
#include <hip/hip_runtime.h>

// Backward-chaining grounder step for MI455X (gfx1250, wave32).
// Kernel A: per-(b,s) ordered wildcard match of 100k facts + 64 rule heads,
//           facts staged through LDS with CDNA5 async-to-LDS DMA
//           (global_load_async_to_lds_b128 / s_wait_asynccnt), double-buffered.
// Kernel B: per-batch ordered top-64 selection + goal substitution/emission.

#define PADV    6000
#define CONSTNO 5000

namespace {
constexpr int F_CNT = 100000;
constexpr int MB_   = 3;
constexpr int B_    = 8;
constexpr int S_    = 64;
constexpr int G_    = 8;
constexpr int KF    = 64;
constexpr int KR    = 16;

constexpr int CHUNK       = 2048;                        // facts per LDS buffer
constexpr int CHUNK_BYTES = CHUNK * 12;                  // 24576 B
constexpr int SEGS        = CHUNK_BYTES / 16;            // 1536 b128 segments
constexpr int NCH         = (F_CNT + CHUNK - 1) / CHUNK; // 49 chunks
constexpr int WS_STRIDE   = 84;  // [nf, nr, fidx[64], ridx[16], pad x2] per (b,s)
}

__device__ __forceinline__ bool is_var(int x) { return (x > CONSTNO) && (x != PADV); }

// Low 32 bits of a generic pointer into LDS == byte offset within the
// workgroup's LDS allocation (ISA 10.2).
__device__ __forceinline__ unsigned lds_lo32(const void* p) {
  return (unsigned)(unsigned long long)p;
}

__global__ __launch_bounds__(256)
void BCGrounder_29394756174278_match(const int* __restrict__ pg,
                                     const int* __restrict__ facts,
                                     const int* __restrict__ heads,
                                     int* __restrict__ ws) {
  __shared__ int ibuf[2][CHUNK_BYTES / 4];  // 2 x 24 KB fact staging buffers
  __shared__ int flist[8][KF];
  __shared__ int rlist[8][KR];

  const int tid  = threadIdx.x;
  const int lane = tid & 31;
  const int wv   = tid >> 5;
  const int pair = blockIdx.x * 8 + wv;     // flattened (b*S + s), one per wave

  for (int k = lane; k < KF; k += 32) flist[wv][k] = 0;
  if (lane < KR) rlist[wv][lane] = 0;

  // ---- query for this wave ----
  const int qb  = pair * G_ * 3;
  const int qp  = pg[qb + 0];
  const int qa0 = pg[qb + 1];
  const int qa1 = pg[qb + 2];
  const bool active = (qp != PADV);
  const bool va0 = is_var(qa0);
  const bool va1 = is_var(qa1);

  // ---- rule-head matching: 64 rules, 2 per lane, ordered via ballot ----
  int rcount = 0;
  #pragma unroll
  for (int half = 0; half < 2; ++half) {
    const int r  = lane + half * 32;
    const int h0 = heads[r * 3 + 0];
    const int h1 = heads[r * 3 + 1];
    const int h2 = heads[r * 3 + 2];
    const bool m = active && (qp == h0) &&
                   (is_var(h1) || va0 || (qa0 == h1)) &&
                   (is_var(h2) || va1 || (qa1 == h2));
    const unsigned bal = __builtin_amdgcn_ballot_w32(m);
    const int rank = rcount + __popc(bal & ((1u << lane) - 1u));
    if (m && rank < KR) rlist[wv][rank] = r;
    rcount += __popc(bal);
  }
  const int nr = rcount > KR ? KR : rcount;

  // ---- fact scan: async double-buffered staging into LDS ----
  const unsigned gbytes = (unsigned)F_CNT * 12u;  // 1,200,000 (16B-aligned)

  auto issue_chunk = [&](int c) {
    const unsigned base    = (unsigned)c * (unsigned)CHUNK_BYTES;
    const unsigned ldsbase = lds_lo32(&ibuf[c & 1][0]);
    #pragma unroll
    for (int k = 0; k < SEGS / 256; ++k) {       // 6 b128 issues per thread
      const int seg = tid + k * 256;
      unsigned off = base + (unsigned)seg * 16u;
      if (off + 16u > gbytes) off = gbytes - 16u; // clamp: keep issue count fixed
      const char* gp = (const char*)facts + off;
      const unsigned dst = ldsbase + (unsigned)seg * 16u;
      asm volatile("global_load_async_to_lds_b128 %0, %1, off"
                   :: "v"(dst), "v"(gp)
                   : "memory");
    }
  };

  issue_chunk(0);
  int fcount = 0;
  for (int c = 0; c < NCH; ++c) {
    if (c + 1 < NCH) {
      issue_chunk(c + 1);
      // 12 async loads outstanding per wave; <=6 means chunk c fully landed
      // (async loads retire in order).
      asm volatile("s_wait_asynccnt 0x6" ::: "memory");
    } else {
      asm volatile("s_wait_asynccnt 0x0" ::: "memory");
    }
    __syncthreads();

    const int rem      = F_CNT - c * CHUNK;
    const int nf_chunk = rem < CHUNK ? rem : CHUNK;
    const int pb       = c & 1;

    if (active && fcount < KF) {
      for (int i0 = 0; i0 < CHUNK; i0 += 32) {   // uniform trip count for ballots
        const int idx = i0 + lane;
        bool m = false;
        if (idx < nf_chunk) {
          const int f0 = ibuf[pb][idx * 3 + 0];
          const int f1 = ibuf[pb][idx * 3 + 1];
          const int f2 = ibuf[pb][idx * 3 + 2];
          m = (qp == f0) && (va0 || (qa0 == f1)) && (va1 || (qa1 == f2));
        }
        const unsigned bal = __builtin_amdgcn_ballot_w32(m);
        const int rank = fcount + __popc(bal & ((1u << lane) - 1u));
        if (m && rank < KF) flist[wv][rank] = c * CHUNK + idx;
        fcount += __popc(bal);                    // wave-uniform
        if (fcount >= KF) break;                  // uniform break
      }
    }
    __syncthreads();  // buffer (c&1) may be overwritten next iteration
  }
  const int nf = fcount > KF ? KF : fcount;

  // ---- write per-query match results to workspace ----
  const int wbase = pair * WS_STRIDE;
  if (lane == 0) { ws[wbase + 0] = nf; ws[wbase + 1] = nr; }
  ws[wbase + 2 + lane]      = flist[wv][lane];
  ws[wbase + 2 + lane + 32] = flist[wv][lane + 32];
  if (lane < KR) ws[wbase + 66 + lane] = rlist[wv][lane];
}

__global__ __launch_bounds__(64)
void BCGrounder_29394756174278_emit(const int* __restrict__ pg,
                                    const int* __restrict__ facts,
                                    const int* __restrict__ heads,
                                    const int* __restrict__ bodies,
                                    const int* __restrict__ lens,
                                    const float* __restrict__ scores,
                                    const int* __restrict__ ws,
                                    float* __restrict__ out) {
  const int b = blockIdx.x;   // 0..7
  const int j = threadIdx.x;  // output slot 0..63
  __shared__ int cnt[S_];
  {
    const int p = b * S_ + j;
    cnt[j] = ws[p * WS_STRIDE + 0] + ws[p * WS_STRIDE + 1];
  }
  __syncthreads();

  // Ordered top-64: first `total` successes in (s, fact-k, rule-k) order.
  int s = -1, local = 0, acc = 0;
  for (int t = 0; t < S_; ++t) {
    const int ct = cnt[t];
    if (s < 0 && j < acc + ct) { s = t; local = j - acc; }
    acc += ct;
  }
  const bool succ = (s >= 0);

  int goals[G_][3];
  int gb[MB_][3];
  #pragma unroll
  for (int g = 0; g < G_; ++g) { goals[g][0] = PADV; goals[g][1] = PADV; goals[g][2] = PADV; }
  #pragma unroll
  for (int t = 0; t < MB_; ++t) { gb[t][0] = PADV; gb[t][1] = PADV; gb[t][2] = PADV; }
  float score = 0.0f;

  if (succ) {
    const int pr = b * S_ + s;
    const int* q = pg + pr * G_ * 3;
    const int qp = q[0], qa0 = q[1], qa1 = q[2];
    const bool va0 = is_var(qa0);
    const bool va1 = is_var(qa1);
    const int nf = ws[pr * WS_STRIDE + 0];

    if (local < nf) {
      // ---- fact expansion: substitute query vars with fact constants ----
      const int fi = ws[pr * WS_STRIDE + 2 + local];
      const int f0 = facts[fi * 3 + 0];
      const int f1 = facts[fi * 3 + 1];
      const int f2 = facts[fi * 3 + 2];
      #pragma unroll
      for (int g = 0; g < G_; ++g) {
        int p, a1, a2;
        if (g == 0) { p = PADV; a1 = PADV; a2 = PADV; }  // remaining: slot0 -> PAD
        else { p = q[g * 3 + 0]; a1 = q[g * 3 + 1]; a2 = q[g * 3 + 2]; }
        if (va0) { if (a1 == qa0) a1 = f1; if (a2 == qa0) a2 = f1; }  // sub qa0 first
        if (va1) { if (a1 == qa1) a1 = f2; if (a2 == qa1) a2 = f2; }  // then qa1
        goals[g][0] = p; goals[g][1] = a1; goals[g][2] = a2;
      }
      gb[0][0] = f0; gb[0][1] = f1; gb[0][2] = f2;
    } else {
      // ---- rule expansion: bind head vars to query constants ----
      const int ri = ws[pr * WS_STRIDE + 66 + (local - nf)];
      const int h1 = heads[ri * 3 + 1];
      const int h2 = heads[ri * 3 + 2];
      const bool c1 = is_var(h1) && !va0 && (qa0 != PADV);
      const bool c2 = is_var(h2) && !va1 && (qa1 != PADV);
      const int len = lens[ri];
      #pragma unroll
      for (int t = 0; t < MB_; ++t) {
        int p  = bodies[(ri * MB_ + t) * 3 + 0];
        int a1 = bodies[(ri * MB_ + t) * 3 + 1];
        int a2 = bodies[(ri * MB_ + t) * 3 + 2];
        if (c1) { if (a1 == h1) a1 = qa0; if (a2 == h1) a2 = qa0; }  // bind h1 first
        if (c2) { if (a1 == h2) a1 = qa1; if (a2 == h2) a2 = qa1; }  // then h2
        if (t >= len) { p = PADV; a1 = PADV; a2 = PADV; }            // mask after bind
        goals[t][0] = p; goals[t][1] = a1; goals[t][2] = a2;
      }
      #pragma unroll
      for (int g = MB_; g < G_; ++g) {     // remaining slots 1..5 appended
        const int src = g - MB_ + 1;
        goals[g][0] = q[src * 3 + 0];
        goals[g][1] = q[src * 3 + 1];
        goals[g][2] = q[src * 3 + 2];
      }
      gb[0][0] = qp; gb[0][1] = qa0; gb[0][2] = qa1;
    }
    score = scores[pr];
  }

  // ---- flat tuple output: goals | gbody | success | scores ----
  const int row = b * S_ + j;
  float* og    = out;
  float* ob    = out + B_ * S_ * G_ * 3;
  float* osucc = ob + B_ * S_ * MB_ * 3;
  float* oscr  = osucc + B_ * S_;
  #pragma unroll
  for (int g = 0; g < G_; ++g)
    #pragma unroll
    for (int k = 0; k < 3; ++k)
      og[(row * G_ + g) * 3 + k] = (float)goals[g][k];
  #pragma unroll
  for (int t = 0; t < MB_; ++t)
    #pragma unroll
    for (int k = 0; k < 3; ++k)
      ob[(row * MB_ + t) * 3 + k] = (float)gb[t][k];
  osucc[row] = succ ? 1.0f : 0.0f;
  oscr[row]  = score;
}

extern "C" void kernel_launch(void* const* d_in, const int* in_sizes, int n_in,
                              void* d_out, int out_size, void* d_ws, size_t ws_size,
                              hipStream_t stream) {
  (void)in_sizes; (void)n_in; (void)out_size; (void)ws_size;
  const int*   pg     = (const int*)d_in[0];   // proof_goals   [B,S,G,3]
  const int*   facts  = (const int*)d_in[1];   // facts_idx     [F,3]
  const int*   heads  = (const int*)d_in[2];   // rules_heads   [R,3]
  const int*   bodies = (const int*)d_in[3];   // rules_bodies  [R,MB,3]
  const int*   lens   = (const int*)d_in[4];   // rule_lens     [R]
  const float* scores = (const float*)d_in[5]; // state_scores  [B,S]
  int*   ws  = (int*)d_ws;                     // 512*84*4 = 168 KB used
  float* out = (float*)d_out;

  BCGrounder_29394756174278_match<<<(B_ * S_) / 8, 256, 0, stream>>>(pg, facts, heads, ws);
  BCGrounder_29394756174278_emit<<<B_, S_, 0, stream>>>(pg, facts, heads, bodies, lens,
                                                        scores, ws, out);
}